// atten_44384192037517
// MI455X (gfx1250) — compile-verified
//
#include <hip/hip_runtime.h>

typedef __attribute__((ext_vector_type(2))) float v2f;
typedef __attribute__((ext_vector_type(4))) float v4f;
typedef __attribute__((ext_vector_type(8))) float v8f;

// Workspace layout (float offsets). Total ~8.4 MB.
#define OFF_WCOMP 0                       // 3*9*9 = 243 composite conv weights
#define OFF_BCOMP 243                     // composite bias
#define OFF_WEFF  256                     // 64x64 collapsed linear
#define OFF_UIMG  4352                    // [4,4096,64] unfolded img (raw -> trans in place)
#define OFF_UFEA  (4352 + 4*4096*64)      // [4,4096,64] unfolded feamap

// ---------- Kernel 1: fold conv2*conv1 -> 9x9 kernel; w2@w1 -> W_eff ----------
__global__ void prep_kernel(const float* __restrict__ w_img1, const float* __restrict__ b_img1,
                            const float* __restrict__ w_img2, const float* __restrict__ b_img2,
                            const float* __restrict__ w1, const float* __restrict__ w2,
                            float* __restrict__ ws) {
  int tid = threadIdx.x;
  if (tid < 243) {                        // w_comp[c][sy][sx]
    int c = tid / 81, r = tid % 81, sy = r / 9, sx = r % 9;
    float acc = 0.f;
    for (int m = 0; m < 64; ++m)
      for (int qy = 0; qy < 3; ++qy) {
        int ry = sy - qy; if ((unsigned)ry >= 7u) continue;
        for (int qx = 0; qx < 3; ++qx) {
          int rx = sx - qx; if ((unsigned)rx >= 7u) continue;
          acc += w_img2[m*9 + qy*3 + qx] * w_img1[((m*3 + c)*7 + ry)*7 + rx];
        }
      }
    ws[OFF_WCOMP + tid] = acc;
  } else if (tid == 243) {                // b_comp
    float acc = b_img2[0];
    for (int m = 0; m < 64; ++m) {
      float s = 0.f;
      for (int q = 0; q < 9; ++q) s += w_img2[m*9 + q];
      acc += b_img1[m] * s;
    }
    ws[OFF_BCOMP] = acc;
  }
  for (int i = 0; i < 16; ++i) {          // W_eff[r][c] = sum_k w2[r,k]*w1[k,c]
    int e = tid + i*256;
    int r = e >> 6, c = e & 63;
    float acc = 0.f;
    for (int k = 0; k < 128; ++k) acc += w2[r*128 + k] * w1[k*64 + c];
    ws[OFF_WEFF + e] = acc;
  }
}

// ---------- Kernel 2: composite 9x9 conv (pad 4) + unfold, one thread/pixel ----------
__global__ void conv9_unfold_kernel(const float* __restrict__ x, const float* __restrict__ ws,
                                    float* __restrict__ u_img_raw) {
  __shared__ float wc[244];
  int tid = threadIdx.x;
  if (tid < 244) wc[tid] = ws[tid];
  __syncthreads();
  int pix = blockIdx.x * 256 + tid;       // 4*512*512 pixels
  int b = pix >> 18, rem = pix & 262143;
  int h = rem >> 9, w = rem & 511;
  float acc = wc[243];
  for (int c = 0; c < 3; ++c) {
    const float* xp = x + ((size_t)(b*3 + c) << 18);
    const float* wk = wc + c*81;
    #pragma unroll
    for (int dy = 0; dy < 9; ++dy) {
      int hy = h + dy - 4;
      if ((unsigned)hy >= 512u) continue;
      const float* xr = xp + (hy << 9);
      #pragma unroll
      for (int dx = 0; dx < 9; ++dx) {
        int wx = w + dx - 4;
        if ((unsigned)wx < 512u) acc += wk[dy*9 + dx] * xr[wx];
      }
    }
  }
  int l = ((h >> 3) << 6) | (w >> 3);
  int p = ((h & 7) << 3) | (w & 7);
  u_img_raw[((((size_t)b << 12) | l) << 6) | p] = acc;
}

// ---------- Kernel 3: exact two-stage conv on the 1-px border (composition != exact there) ----------
__global__ void border_fix_kernel(const float* __restrict__ x,
                                  const float* __restrict__ w_img1, const float* __restrict__ b_img1,
                                  const float* __restrict__ w_img2, const float* __restrict__ b_img2,
                                  float* __restrict__ u_img_raw) {
  int wg = blockIdx.x;
  int b = wg / 2044, i = wg % 2044;
  int h, w;
  if (i < 512)       { h = 0;   w = i; }
  else if (i < 1024) { h = 511; w = i - 512; }
  else { int r = i - 1024; h = 1 + (r >> 1); w = (r & 1) ? 511 : 0; }
  int m = threadIdx.x;                    // channel 0..63
  const float* xb = x + (((size_t)b * 3) << 18);
  float s = 0.f;
  for (int qy = 0; qy < 3; ++qy) {
    int ty = h + qy - 1; if ((unsigned)ty >= 512u) continue;
    for (int qx = 0; qx < 3; ++qx) {
      int tx = w + qx - 1; if ((unsigned)tx >= 512u) continue;
      float y = b_img1[m];                // y[m, ty, tx] from zero-padded x
      for (int c = 0; c < 3; ++c) {
        const float* xc = xb + ((size_t)c << 18);
        const float* wk = w_img1 + (m*3 + c)*49;
        for (int ry = 0; ry < 7; ++ry) {
          int hy = ty + ry - 3; if ((unsigned)hy >= 512u) continue;
          const float* xr = xc + (hy << 9);
          for (int rx = 0; rx < 7; ++rx) {
            int wx = tx + rx - 3;
            if ((unsigned)wx < 512u) y += wk[ry*7 + rx] * xr[wx];
          }
        }
      }
      s += w_img2[m*9 + qy*3 + qx] * y;
    }
  }
  __shared__ float red[64];
  red[m] = s;
  __syncthreads();
  if (m == 0) {
    float z = b_img2[0];
    for (int j = 0; j < 64; ++j) z += red[j];
    int l = ((h >> 3) << 6) | (w >> 3);
    int p = ((h & 7) << 3) | (w & 7);
    u_img_raw[((((size_t)b << 12) | l) << 6) | p] = z;
  }
}

// ---------- Kernel 4: 1x1 conv on fea (/4 folded) + unfold. Bandwidth bound: 268 MB read ----------
__global__ void feamap_unfold_kernel(const float* __restrict__ fea, const float* __restrict__ w_fea,
                                     const float* __restrict__ b_fea, float* __restrict__ u_fea_raw) {
  __shared__ float wf[64];
  int tid = threadIdx.x;
  if (tid < 64) wf[tid] = w_fea[tid];
  __syncthreads();
  int pix = blockIdx.x * 256 + tid;
  int b = pix >> 18, rem = pix & 262143;
  int h = rem >> 9, w = rem & 511;
  const float* fb = fea + (((size_t)b * 64) << 18) + (h << 9) + w;
  float acc = b_fea[0];
  #pragma unroll 8
  for (int c = 0; c < 64; ++c) acc += wf[c] * fb[(size_t)c << 18];
  int l = ((h >> 3) << 6) | (w >> 3);
  int p = ((h & 7) << 3) | (w & 7);
  u_fea_raw[((((size_t)b << 12) | l) << 6) | p] = acc * 0.25f;
}

// ---------- Kernel 5: u = relu(u_raw @ W_eff^T) * 0.125, in place. WMMA f32 16x16x4 ----------
// One wave per 16-row tile; 2048 tiles (img tiles 0..1023, fea 1024..2047).
__global__ void trans_kernel(float* __restrict__ ws) {
  __shared__ float W[64*68];              // stride 68: conflict-free b-fragment reads
  int tid = threadIdx.x;
  const float* Weff = ws + OFF_WEFF;
  for (int i = 0; i < 16; ++i) {
    int e = tid + i*256;
    W[(e >> 6)*68 + (e & 63)] = Weff[e];
  }
  __syncthreads();
  int wave = tid >> 5, lane = tid & 31;
  int t = blockIdx.x * 8 + wave;
  float* U = ws + (t < 1024 ? OFF_UIMG : OFF_UFEA);
  int row0 = (t & 1023) * 16;
  int mr = lane & 15, kh = lane >> 4;
  const float* arow = U + (size_t)(row0 + mr)*64 + 2*kh;   // A frag: lane<16 K0/1, lane>=16 K2/3
  v2f a[16];
  #pragma unroll
  for (int kk = 0; kk < 16; ++kk) a[kk] = *(const v2f*)(arow + 4*kk);
  v8f acc[4] = {};
  #pragma unroll
  for (int kk = 0; kk < 16; ++kk) {
    #pragma unroll
    for (int n = 0; n < 4; ++n) {
      v2f bv = *(const v2f*)(&W[(n*16 + mr)*68 + 4*kk + 2*kh]);
      acc[n] = __builtin_amdgcn_wmma_f32_16x16x4_f32(false, a[kk], false, bv,
                                                     (short)0, acc[n], false, false);
    }
  }
  #pragma unroll
  for (int n = 0; n < 4; ++n) {
    #pragma unroll
    for (int v = 0; v < 8; ++v)           // C/D: lane<16 -> M=v, lane>=16 -> M=v+8
      U[(size_t)(row0 + v + 8*kh)*64 + n*16 + mr] = fmaxf(acc[n][v], 0.0f) * 0.125f;
  }
}

// ---------- Kernel 6: att[b,l,m] = sum_p Uimg[l,p]*Ufea[m,p] (scale already folded) ----------
// 128x128 tile per 8-wave block; Ufea panel staged in LDS; 128 v_wmma_f32_16x16x4 per wave.
__global__ void attn_kernel(const float* __restrict__ ws, float* __restrict__ out) {
  __shared__ __align__(16) float Bs[128*68];
  int tid = threadIdx.x;
  int col0 = blockIdx.x * 128, row0 = blockIdx.y * 128, b = blockIdx.z;
  const float* Ui = ws + OFF_UIMG + (size_t)b * (4096*64);
  const float* Uf = ws + OFF_UFEA + (size_t)b * (4096*64);
  #pragma unroll
  for (int i = 0; i < 8; ++i) {           // 2048 float4 = 128 rows x 64 floats
    int e = tid + i*256;
    int r = e >> 4, kq = e & 15;
    v4f val = *(const v4f*)(Uf + (size_t)(col0 + r)*64 + kq*4);
    *(v4f*)(&Bs[r*68 + kq*4]) = val;      // 16B-aligned ds_store_b128
  }
  __syncthreads();
  int wave = tid >> 5, lane = tid & 31;
  int mr = lane & 15, kh = lane >> 4;
  const float* arow = Ui + (size_t)(row0 + wave*16 + mr)*64 + 2*kh;
  v2f a[16];
  #pragma unroll
  for (int kk = 0; kk < 16; ++kk) a[kk] = *(const v2f*)(arow + 4*kk);
  v8f acc[8] = {};
  #pragma unroll
  for (int kk = 0; kk < 16; ++kk) {
    #pragma unroll
    for (int nt = 0; nt < 8; ++nt) {
      v2f bv = *(const v2f*)(&Bs[(nt*16 + mr)*68 + 4*kk + 2*kh]);
      acc[nt] = __builtin_amdgcn_wmma_f32_16x16x4_f32(false, a[kk], false, bv,
                                                      (short)0, acc[nt], false, false);
    }
  }
  float* ob = out + (size_t)b * (4096u*4096u);
  int lbase = row0 + wave*16 + 8*kh;
  #pragma unroll
  for (int nt = 0; nt < 8; ++nt) {
    int m = col0 + nt*16 + mr;
    #pragma unroll
    for (int v = 0; v < 8; ++v)
      ob[(size_t)(lbase + v)*4096 + m] = acc[nt][v];
  }
}

extern "C" void kernel_launch(void* const* d_in, const int* in_sizes, int n_in,
                              void* d_out, int out_size, void* d_ws, size_t ws_size,
                              hipStream_t stream) {
  const float* x      = (const float*)d_in[0];
  const float* fea    = (const float*)d_in[1];
  const float* w_img1 = (const float*)d_in[2];
  const float* b_img1 = (const float*)d_in[3];
  const float* w_img2 = (const float*)d_in[4];
  const float* b_img2 = (const float*)d_in[5];
  const float* w_fea  = (const float*)d_in[6];
  const float* b_fea  = (const float*)d_in[7];
  const float* w1     = (const float*)d_in[8];
  const float* w2     = (const float*)d_in[9];
  float* out = (float*)d_out;
  float* ws  = (float*)d_ws;

  prep_kernel<<<1, 256, 0, stream>>>(w_img1, b_img1, w_img2, b_img2, w1, w2, ws);
  conv9_unfold_kernel<<<4096, 256, 0, stream>>>(x, ws, ws + OFF_UIMG);
  border_fix_kernel<<<4*2044, 64, 0, stream>>>(x, w_img1, b_img1, w_img2, b_img2, ws + OFF_UIMG);
  feamap_unfold_kernel<<<4096, 256, 0, stream>>>(fea, w_fea, b_fea, ws + OFF_UFEA);
  trans_kernel<<<256, 256, 0, stream>>>(ws);
  attn_kernel<<<dim3(32, 32, 4), 256, 0, stream>>>(ws, out);
}